// NodeAttention_87591563034730
// MI455X (gfx1250) — compile-verified
//
#include <hip/hip_runtime.h>
#include <hip/hip_bf16.h>

// ---------------------------------------------------------------------------
// NodeAttention (graph transformer edge-softmax attention) for gfx1250.
// Dense GEMMs use V_WMMA_F32_16X16X4_F32 (native fp32 WMMA, wave32).
// Edge-MLP weights are staged into LDS via GLOBAL_LOAD_ASYNC_TO_LDS_B128
// (ASYNCcnt path). Gathers/scatters ride the 192MB L2 (Q/K/V are 25.6MB).
// ---------------------------------------------------------------------------

typedef float v2f __attribute__((ext_vector_type(2)));
typedef float v8f __attribute__((ext_vector_type(8)));

#define DIM       128
#define NHEAD     8
#define DK        16
#define EDGE_DIM  64
#define LDSPITCH  66   // 64 + 2 pad: conflict-free b64 column reads, 8B aligned

// ---- CDNA5 async global->LDS copy (16B per active lane), ASYNCcnt tracked
__device__ __forceinline__ void async_b128(void* lds, const float* g) {
    unsigned int l = (unsigned int)(size_t)lds;   // low 32 bits == LDS offset
    asm volatile("global_load_async_to_lds_b128 %0, %1, off"
                 :: "v"(l), "v"(g) : "memory");
}
__device__ __forceinline__ void wait_async0() {
    asm volatile("s_wait_asynccnt 0x0" ::: "memory");
}

// ---- order-preserving float <-> uint encoding for atomicMax-based segment max
__device__ __forceinline__ unsigned int fenc(float f) {
    unsigned int b = __float_as_uint(f);
    return (b & 0x80000000u) ? ~b : (b | 0x80000000u);
}
__device__ __forceinline__ float fdec(unsigned int k) {
    unsigned int b = (k & 0x80000000u) ? (k ^ 0x80000000u) : ~k;
    return __uint_as_float(b);
}
// encoded -FLT_MAX (bits 0xFF7FFFFF -> key ~bits = 0x00800000)
#define NEG_MAX_KEY 0x00800000u

// ---------------------------------------------------------------------------
// Y[N,128] = X[N,128] @ W^T + b   (W is torch-style [out,in] row-major)
// One wave per 16x16 output tile; blockDim = (32,8): 8 waves cover 128 cols.
// Weight reuse across the 3125 blocks is served by L2 (64KB matrix).
// ---------------------------------------------------------------------------
__global__ __launch_bounds__(256) void proj128_kernel(
    const float* __restrict__ X, const float* __restrict__ W,
    const float* __restrict__ b, float* __restrict__ Y, int N)
{
    const int lane = threadIdx.x;
    const int half = lane >> 4;     // 0: K pair {0,1}; 1: K pair {2,3}
    const int l15  = lane & 15;
    const int m0   = blockIdx.x * 16;
    const int n0   = threadIdx.y * 16;

    v8f c = {};
    #pragma unroll 4
    for (int k0 = 0; k0 < DIM; k0 += 4) {
        const int ka = k0 + 2 * half;
        v2f a  = *(const v2f*)(X + (m0 + l15) * DIM + ka);
        v2f bm = *(const v2f*)(W + (n0 + l15) * DIM + ka);
        c = __builtin_amdgcn_wmma_f32_16x16x4_f32(false, a, false, bm,
                                                  (short)0, c, false, false);
    }
    const float bb = b[n0 + l15];
    #pragma unroll
    for (int r = 0; r < 8; ++r) {
        const int row = m0 + r + 8 * half;          // C/D layout: lanes16-31 -> M+8
        Y[row * DIM + n0 + l15] = c[r] + bb;
    }
}

// ---------------------------------------------------------------------------
// Edge-bias MLP:  bias[E,8] = silu(EA[E,64] @ W1^T + b1) @ W2^T + b2
// One wave per 16-edge tile, 8 tiles per block. W1/W2 staged once per block
// into padded LDS via async b128 copies; h transposed C->A layout through LDS.
// ---------------------------------------------------------------------------
__global__ __launch_bounds__(256) void edge_mlp_kernel(
    const float* __restrict__ EA, const float* __restrict__ W1,
    const float* __restrict__ b1, const float* __restrict__ W2,
    const float* __restrict__ b2, float* __restrict__ biasOut, int E)
{
    __shared__ float w1s[EDGE_DIM][LDSPITCH];     // 16.9 KB
    __shared__ float w2s[NHEAD][LDSPITCH];        //  2.1 KB
    __shared__ float hbuf[8][16][LDSPITCH];       // 33.8 KB

    const int lane = threadIdx.x;
    const int half = lane >> 4;
    const int l15  = lane & 15;
    const int wv   = threadIdx.y;                 // wave id within block
    const int tid  = wv * 32 + lane;              // 0..255
    const int e0   = (blockIdx.x * 8 + wv) * 16;  // first edge of this tile

    // ---- stage weights to LDS: 4096+512 floats as 16B async copies
    #pragma unroll
    for (int it = 0; it < 4; ++it) {
        const int f   = it * 256 + tid;           // float4 index within W1
        const int row = f >> 4, cg = f & 15;
        async_b128(&w1s[row][cg * 4], W1 + f * 4);
    }
    if (tid < 128) {                              // W2: 8x64 = 128 float4
        const int row = tid >> 4, cg = tid & 15;
        async_b128(&w2s[row][cg * 4], W2 + tid * 4);
    }
    // keep the 205MB edge_attr stream ahead of compute
    __builtin_prefetch(EA + (size_t)(e0 + 128) * EDGE_DIM, 0, 1);
    wait_async0();
    __syncthreads();

    // ---- stage 1: h = silu(EA @ W1^T + b1), four 16x16 tiles wide
    v8f acc[4] = {};
    #pragma unroll 2
    for (int k0 = 0; k0 < EDGE_DIM; k0 += 4) {
        const int ka = k0 + 2 * half;
        v2f a = *(const v2f*)(EA + (e0 + l15) * EDGE_DIM + ka);
        #pragma unroll
        for (int t = 0; t < 4; ++t) {
            v2f bm = *(const v2f*)&w1s[t * 16 + l15][ka];
            acc[t] = __builtin_amdgcn_wmma_f32_16x16x4_f32(false, a, false, bm,
                                                           (short)0, acc[t],
                                                           false, false);
        }
    }
    #pragma unroll
    for (int t = 0; t < 4; ++t) {
        const float bb = b1[t * 16 + l15];
        #pragma unroll
        for (int r = 0; r < 8; ++r) {
            const float xv = acc[t][r] + bb;
            const float s  = xv / (1.0f + __expf(-xv));     // SiLU
            hbuf[wv][r + 8 * half][t * 16 + l15] = s;
        }
    }
    __syncthreads();

    // ---- stage 2: bias = h @ W2^T + b2  (W2 is [8,64]; pad cols 8..15 = 0)
    v8f c2 = {};
    #pragma unroll 2
    for (int k0 = 0; k0 < EDGE_DIM; k0 += 4) {
        const int ka = k0 + 2 * half;
        v2f a = *(const v2f*)&hbuf[wv][l15][ka];
        v2f bm;
        if (l15 < NHEAD) bm = *(const v2f*)&w2s[l15][ka];
        else             { bm.x = 0.0f; bm.y = 0.0f; }
        c2 = __builtin_amdgcn_wmma_f32_16x16x4_f32(false, a, false, bm,
                                                   (short)0, c2, false, false);
    }
    if (l15 < NHEAD) {
        const float bb = b2[l15];
        #pragma unroll
        for (int r = 0; r < 8; ++r)
            biasOut[(e0 + r + 8 * half) * NHEAD + l15] = c2[r] + bb;
    }
}

// ---------------------------------------------------------------------------
// Init: zero attn accumulator + denom, seed segment-max keys with -FLT_MAX.
// ---------------------------------------------------------------------------
__global__ void init_kernel(unsigned int* __restrict__ smaxKey,
                            float* __restrict__ denom,
                            float* __restrict__ pre, int nNodes)
{
    const int tid = blockIdx.x * blockDim.x + threadIdx.x;
    if (tid < nNodes * DIM)   pre[tid] = 0.0f;
    if (tid < nNodes * NHEAD) { smaxKey[tid] = NEG_MAX_KEY; denom[tid] = 0.0f; }
}

// ---------------------------------------------------------------------------
// Per-(edge,head): score = q[j].k[i]/sqrt(dk) + bias; atomicMax segment max.
// Q/K are 25.6MB each -> gathers hit L2.
// ---------------------------------------------------------------------------
__global__ void scores_kernel(const float* __restrict__ Q,
                              const float* __restrict__ K,
                              const float* __restrict__ bias,
                              const int* __restrict__ src,
                              const int* __restrict__ dst,
                              float* __restrict__ sc,
                              unsigned int* __restrict__ smaxKey, int E)
{
    const int tid = blockIdx.x * blockDim.x + threadIdx.x;
    if (tid >= E * NHEAD) return;
    const int e = tid >> 3, h = tid & 7;
    const int i = src[e], j = dst[e];

    const float4* qp = (const float4*)(Q + j * DIM + h * DK);
    const float4* kp = (const float4*)(K + i * DIM + h * DK);
    float d = 0.0f;
    #pragma unroll
    for (int t = 0; t < 4; ++t) {
        float4 qv = qp[t], kv = kp[t];
        d += qv.x * kv.x + qv.y * kv.y + qv.z * kv.z + qv.w * kv.w;
    }
    const float s = d * 0.25f /* 1/sqrt(16) */ + bias[tid];
    sc[tid] = s;
    atomicMax(&smaxKey[j * NHEAD + h], fenc(s));
}

// ---------------------------------------------------------------------------
// ex = exp(score - segmax[j]); accumulate segment sums.
// ---------------------------------------------------------------------------
__global__ void expsum_kernel(float* __restrict__ sc,
                              const int* __restrict__ dst,
                              const unsigned int* __restrict__ smaxKey,
                              float* __restrict__ denom, int E)
{
    const int tid = blockIdx.x * blockDim.x + threadIdx.x;
    if (tid >= E * NHEAD) return;
    const int e = tid >> 3, h = tid & 7;
    const int j = dst[e];
    const float m  = fdec(smaxKey[j * NHEAD + h]);
    const float ex = __expf(sc[tid] - m);
    sc[tid] = ex;
    atomicAdd(&denom[j * NHEAD + h], ex);
}

// ---------------------------------------------------------------------------
// alpha = ex/denom[j]; scatter-add alpha * v[i] into attn accumulator (L2).
// ---------------------------------------------------------------------------
__global__ void scatter_kernel(const float* __restrict__ sc,
                               const float* __restrict__ denom,
                               const float* __restrict__ V,
                               const int* __restrict__ src,
                               const int* __restrict__ dst,
                               float* __restrict__ pre, int E)
{
    const int tid = blockIdx.x * blockDim.x + threadIdx.x;
    if (tid >= E * NHEAD) return;
    const int e = tid >> 3, h = tid & 7;
    const int i = src[e], j = dst[e];
    const float alpha = sc[tid] / denom[j * NHEAD + h];

    const float4* vp = (const float4*)(V + i * DIM + h * DK);
    float* op = pre + j * DIM + h * DK;
    #pragma unroll
    for (int t = 0; t < 4; ++t) {
        float4 vv = vp[t];
        atomicAdd(op + 4 * t + 0, alpha * vv.x);
        atomicAdd(op + 4 * t + 1, alpha * vv.y);
        atomicAdd(op + 4 * t + 2, alpha * vv.z);
        atomicAdd(op + 4 * t + 3, alpha * vv.w);
    }
}

// ---------------------------------------------------------------------------
// Host-side orchestration (graph-capture safe: launches only).
// ---------------------------------------------------------------------------
extern "C" void kernel_launch(void* const* d_in, const int* in_sizes, int n_in,
                              void* d_out, int out_size, void* d_ws, size_t ws_size,
                              hipStream_t stream)
{
    (void)n_in; (void)out_size; (void)ws_size;

    const float* x    = (const float*)d_in[0];
    const int*   eidx = (const int*)  d_in[1];   // [2,E]
    const float* ea   = (const float*)d_in[2];
    const float* WQ   = (const float*)d_in[3];
    const float* bQ   = (const float*)d_in[4];
    const float* WK   = (const float*)d_in[5];
    const float* bK   = (const float*)d_in[6];
    const float* WV   = (const float*)d_in[7];
    const float* bV   = (const float*)d_in[8];
    const float* WO   = (const float*)d_in[9];
    const float* bO   = (const float*)d_in[10];
    const float* We1  = (const float*)d_in[11];
    const float* be1  = (const float*)d_in[12];
    const float* We2  = (const float*)d_in[13];
    const float* be2  = (const float*)d_in[14];

    const int N = in_sizes[0] / DIM;        // 50000
    const int E = in_sizes[2] / EDGE_DIM;   // 800000
    const int* src = eidx;                  // edge_index[0]
    const int* dst = eidx + E;              // edge_index[1]

    // -------- workspace carve-up (fp32 elements) --------
    float* fws  = (float*)d_ws;
    float* Qb   = fws;
    float* Kb   = Qb  + (size_t)N * DIM;
    float* Vb   = Kb  + (size_t)N * DIM;
    float* PRE  = Vb  + (size_t)N * DIM;
    float* BIAS = PRE + (size_t)N * DIM;
    float* SC   = BIAS + (size_t)E * NHEAD;
    float* DEN  = SC  + (size_t)E * NHEAD;
    unsigned int* KEY = (unsigned int*)(DEN + (size_t)N * NHEAD);

    const dim3 wblk(32, 8);                 // 8 wave32 per block

    // Q/K/V projections (WMMA fp32)
    proj128_kernel<<<dim3(N / 16), wblk, 0, stream>>>(x, WQ, bQ, Qb, N);
    proj128_kernel<<<dim3(N / 16), wblk, 0, stream>>>(x, WK, bK, Kb, N);
    proj128_kernel<<<dim3(N / 16), wblk, 0, stream>>>(x, WV, bV, Vb, N);

    // edge-bias MLP (WMMA fp32 + async-LDS weight staging)
    edge_mlp_kernel<<<dim3(E / (16 * 8)), wblk, 0, stream>>>(
        ea, We1, be1, We2, be2, BIAS, E);

    // reset accumulators / segment-max keys
    init_kernel<<<dim3((N * DIM + 255) / 256), dim3(256), 0, stream>>>(
        KEY, DEN, PRE, N);

    const int EH = E * NHEAD;
    scores_kernel<<<dim3((EH + 255) / 256), dim3(256), 0, stream>>>(
        Qb, Kb, BIAS, src, dst, SC, KEY, E);
    expsum_kernel<<<dim3((EH + 255) / 256), dim3(256), 0, stream>>>(
        SC, dst, KEY, DEN, E);
    scatter_kernel<<<dim3((EH + 255) / 256), dim3(256), 0, stream>>>(
        SC, DEN, Vb, src, dst, PRE, E);

    // output projection straight into d_out
    proj128_kernel<<<dim3(N / 16), wblk, 0, stream>>>(
        PRE, WO, bO, (float*)d_out, N);
}